// EnhancedPositionalEncoding_1168231104868
// MI455X (gfx1250) — compile-verified
//
#include <hip/hip_runtime.h>
#include <math.h>

#define NB 4
#define NP 4096
#define KNN 16
#define TILE 256

typedef float v2f __attribute__((ext_vector_type(2)));
typedef float v8f __attribute__((ext_vector_type(8)));

// Exact-f32 WMMA: D(16x16) += A(16x4) * B(4x16), CDNA5 V_WMMA_F32_16X16X4_F32
__device__ __forceinline__ v8f wmma4(v2f a, v2f b, v8f c) {
  return __builtin_amdgcn_wmma_f32_16x16x4_f32(false, a, false, b, (short)0, c,
                                               false, false);
}

// ---------------------------------------------------------------------------
// Kernel A: brute-force kNN (top-16 smallest d2) + rel_pos gather.
// One thread per query point; candidates staged through LDS in 256-pt tiles.
// d2 = sq_q + sq_c - 2*dot  (same arithmetic form as the reference, so the
// self point gets exactly 0 and is always kept).
// ---------------------------------------------------------------------------
__global__ void knn_relpos_kernel(const float* __restrict__ xyz,
                                  float* __restrict__ relpos) {
  __shared__ float sx[TILE], sy[TILE], sz[TILE], ssq[TILE];
  const int b = blockIdx.x / (NP / TILE);
  const int q = (blockIdx.x % (NP / TILE)) * TILE + threadIdx.x;
  const float* base = xyz + (size_t)b * NP * 3;

  const float qx = base[q * 3 + 0];
  const float qy = base[q * 3 + 1];
  const float qz = base[q * 3 + 2];
  const float qsq = qx * qx + qy * qy + qz * qz;

  float bd[KNN];
  int bi[KNN];
#pragma unroll
  for (int j = 0; j < KNN; ++j) { bd[j] = 3.0e38f; bi[j] = 0; }
  float worst = 3.0e38f;
  int worstSlot = 0;

  for (int t = 0; t < NP; t += TILE) {
    __syncthreads();
    {
      const int c = t + threadIdx.x;
      const float x = base[c * 3 + 0];
      const float y = base[c * 3 + 1];
      const float z = base[c * 3 + 2];
      sx[threadIdx.x] = x;
      sy[threadIdx.x] = y;
      sz[threadIdx.x] = z;
      ssq[threadIdx.x] = x * x + y * y + z * z;
    }
    __syncthreads();
    for (int i = 0; i < TILE; ++i) {
      const float dot = qx * sx[i] + qy * sy[i] + qz * sz[i];
      const float d2 = qsq + ssq[i] - 2.0f * dot;
      if (d2 < worst) {
        const int cand = t + i;
#pragma unroll
        for (int j = 0; j < KNN; ++j)
          if (j == worstSlot) { bd[j] = d2; bi[j] = cand; }
        worst = bd[0];
        worstSlot = 0;
#pragma unroll
        for (int j = 1; j < KNN; ++j)
          if (bd[j] > worst) { worst = bd[j]; worstSlot = j; }
      }
    }
  }

  float* rp = relpos + ((size_t)(b * NP + q)) * (KNN * 3);
#pragma unroll
  for (int j = 0; j < KNN; ++j) {
    const int ni = bi[j];
    rp[j * 3 + 0] = base[ni * 3 + 0] - qx;
    rp[j * 3 + 1] = base[ni * 3 + 1] - qy;
    rp[j * 3 + 2] = base[ni * 3 + 2] - qz;
  }
}

// ---------------------------------------------------------------------------
// Kernel B: relative encoding MLP. One wave32 per point.
// A-matrix rows = 16 neighbors, features = 28 cols -> 7 WMMA K-steps;
// BN+ReLU folded per output channel (each lane owns one column of C/D);
// layer2 16x16 -> 4 WMMA K-steps; mean over the 16 rows (K axis) + b2.
// ---------------------------------------------------------------------------
__global__ void rel_mlp_kernel(const float* __restrict__ relpos,
                               const float* __restrict__ W1,
                               const float* __restrict__ b1,
                               const float* __restrict__ g,
                               const float* __restrict__ be,
                               const float* __restrict__ m,
                               const float* __restrict__ v,
                               const float* __restrict__ W2,
                               const float* __restrict__ b2,
                               float* __restrict__ out) {
  __shared__ float feat[8][16][32];  // per-wave 16x28 feature tile (pitch 32)
  const int lane = threadIdx.x & 31;
  const int wave = threadIdx.x >> 5;
  const int p = blockIdx.x * 8 + wave;  // global point id, 0..16383
  const int half = lane >> 4;
  const int mrow = lane & 15;  // A row (neighbor k) / B,C column (channel)

  // Phase 1: lanes 0-15 each build the 28-dim feature for neighbor k = lane.
  if (half == 0) {
    const float* rp = relpos + (size_t)p * (KNN * 3) + lane * 3;
    const float x = rp[0], y = rp[1], z = rp[2];
    const float d = sqrtf(x * x + y * y + z * z);
    const float inv = 1.0f / (d + 1e-8f);
    float* f = feat[wave][lane];
#pragma unroll
    for (int fi = 0; fi < 4; ++fi) {
      const float fr = (float)(1 << fi);  // 1,2,4,8
      f[fi * 6 + 0] = __sinf(fr * x);
      f[fi * 6 + 1] = __sinf(fr * y);
      f[fi * 6 + 2] = __sinf(fr * z);
      f[fi * 6 + 3] = __cosf(fr * x);
      f[fi * 6 + 4] = __cosf(fr * y);
      f[fi * 6 + 5] = __cosf(fr * z);
    }
    f[24] = d;
    f[25] = x * inv;
    f[26] = y * inv;
    f[27] = z * inv;
    f[28] = 0.f; f[29] = 0.f; f[30] = 0.f; f[31] = 0.f;
  }
  __syncthreads();

  // Layer 1: D = feat(16x28) x W1^T(28x16), 7 exact-f32 WMMAs.
  v8f acc = {0.f, 0.f, 0.f, 0.f, 0.f, 0.f, 0.f, 0.f};
#pragma unroll
  for (int kk = 0; kk < 7; ++kk) {
    const int k0 = 4 * kk + 2 * half;
    v2f a = {feat[wave][mrow][k0], feat[wave][mrow][k0 + 1]};
    v2f bf = {W1[mrow * 28 + k0], W1[mrow * 28 + k0 + 1]};
    acc = wmma4(a, bf, acc);
  }

  // BN (eval) + bias fold + ReLU. Each lane owns output channel `mrow`.
  const float sc = g[mrow] * rsqrtf(v[mrow] + 1e-5f);
  const float tt = (b1[mrow] - m[mrow]) * sc + be[mrow];
  float h[8];
#pragma unroll
  for (int j = 0; j < 8; ++j) {
    const float x = acc[j] * sc + tt;
    h[j] = x > 0.f ? x : 0.f;
  }

  __syncthreads();  // WAR: layer-1 fragment reads vs h stores
#pragma unroll
  for (int j = 0; j < 8; ++j) feat[wave][j + 8 * half][mrow] = h[j];
  __syncthreads();

  // Layer 2: D2 = h(16x16) x W2^T(16x16), 4 WMMAs.
  v8f acc2 = {0.f, 0.f, 0.f, 0.f, 0.f, 0.f, 0.f, 0.f};
#pragma unroll
  for (int kk = 0; kk < 4; ++kk) {
    const int k0 = 4 * kk + 2 * half;
    v2f a = {feat[wave][mrow][k0], feat[wave][mrow][k0 + 1]};
    v2f bf = {W2[mrow * 16 + k0], W2[mrow * 16 + k0 + 1]};
    acc2 = wmma4(a, bf, acc2);
  }

  // Mean over K (rows): lane holds rows j+8*half of column `mrow`.
  float s = 0.f;
#pragma unroll
  for (int j = 0; j < 8; ++j) s += acc2[j];
  s += __shfl_xor(s, 16, 32);
  if (half == 0) {
    const float val = s * (1.0f / 16.0f) + b2[mrow];
    const int bb = p >> 12, nn = p & (NP - 1);
    out[((size_t)(bb * 32 + mrow)) * NP + nn] = val;  // channels 0..15
  }
}

// ---------------------------------------------------------------------------
// Kernel C1: per-point structure features (22-dim, stored pitch-24, zero pad).
// ---------------------------------------------------------------------------
__global__ void struct_feat_kernel(const float* __restrict__ relpos,
                                   float* __restrict__ sf) {
  const int p = blockIdx.x * blockDim.x + threadIdx.x;
  const float* rp = relpos + (size_t)p * (KNN * 3);

  float rx[KNN], ry[KNN], rz[KNN];
  float sx = 0.f, sy = 0.f, sz = 0.f;
  float cxx = 0.f, cxy = 0.f, cxz = 0.f, cyy = 0.f, cyz = 0.f, czz = 0.f;
  float nxs = 0.f, nys = 0.f, nzs = 0.f;
#pragma unroll
  for (int k = 0; k < KNN; ++k) {
    const float x = rp[k * 3 + 0], y = rp[k * 3 + 1], z = rp[k * 3 + 2];
    rx[k] = x; ry[k] = y; rz[k] = z;
    sx += x; sy += y; sz += z;
    cxx += x * x; cxy += x * y; cxz += x * z;
    cyy += y * y; cyz += y * z; czz += z * z;
    const float d = sqrtf(x * x + y * y + z * z);
    const float inv = 1.0f / (d + 1e-8f);
    nxs += x * inv; nys += y * inv; nzs += z * inv;
  }
  const float inv15 = 1.0f / 15.0f;
  const float c00 = cxx * inv15, c01 = cxy * inv15, c02 = cxz * inv15;
  const float c11 = cyy * inv15, c12 = cyz * inv15, c22 = czz * inv15;
  const float mx = sx * (1.0f / 16.0f), my = sy * (1.0f / 16.0f),
              mz = sz * (1.0f / 16.0f);

  float vx = 0.f, vy = 0.f, vz = 0.f, dmax = 0.f, dmin = 3.0e38f;
#pragma unroll
  for (int k = 0; k < KNN; ++k) {
    const float ex = rx[k] - mx, ey = ry[k] - my, ez = rz[k] - mz;
    vx += ex * ex; vy += ey * ey; vz += ez * ez;
    const float dc = sqrtf(ex * ex + ey * ey + ez * ez);
    dmax = fmaxf(dmax, dc);
    dmin = fminf(dmin, dc);
  }
  const float stdx = sqrtf(vx * inv15);
  const float stdy = sqrtf(vy * inv15);
  const float stdz = sqrtf(vz * inv15);
  const float local_radius = dmax;
  const float density = 16.0f / (dmax + 1e-8f);
  // mean of sorted adjacent diffs telescopes to (max - min)/15
  const float mean_curv = (dmax - dmin) * inv15;
  const float dir_cons = (nxs * nxs + nys * nys + nzs * nzs) * (1.0f / 256.0f);

  // Closed-form eigenvalues of symmetric 3x3 covariance (ascending).
  const float qv = (c00 + c11 + c22) * (1.0f / 3.0f);
  const float p1 = c01 * c01 + c02 * c02 + c12 * c12;
  const float a0 = c00 - qv, a1 = c11 - qv, a2 = c22 - qv;
  const float p2 = a0 * a0 + a1 * a1 + a2 * a2 + 2.0f * p1;
  float ev0, ev1, ev2;  // ascending
  if (p2 < 1e-30f) {
    ev0 = ev1 = ev2 = qv;
  } else {
    const float pp = sqrtf(p2 * (1.0f / 6.0f));
    const float ip = 1.0f / pp;
    const float b00 = a0 * ip, b11 = a1 * ip, b22 = a2 * ip;
    const float b01 = c01 * ip, b02 = c02 * ip, b12 = c12 * ip;
    float detB = b00 * (b11 * b22 - b12 * b12) -
                 b01 * (b01 * b22 - b12 * b02) +
                 b02 * (b01 * b12 - b11 * b02);
    float r = 0.5f * detB;
    r = fminf(1.0f, fmaxf(-1.0f, r));
    const float phi = acosf(r) * (1.0f / 3.0f);
    const float e1 = qv + 2.0f * pp * cosf(phi);                       // largest
    const float e3 = qv + 2.0f * pp * cosf(phi + 2.0943951023931953f); // smallest
    const float e2 = 3.0f * qv - e1 - e3;
    ev0 = e3; ev1 = e2; ev2 = e1;
  }
  const float den = ev0 + 1e-8f;
  const float aniso = (ev0 - ev2) / den;
  const float plan = (ev1 - ev2) / den;
  const float spher = ev2 / den;

  float* o = sf + (size_t)p * 24;
  o[0] = c00; o[1] = c01; o[2] = c02;
  o[3] = c01; o[4] = c11; o[5] = c12;
  o[6] = c02; o[7] = c12; o[8] = c22;
  o[9] = aniso; o[10] = plan; o[11] = spher;
  o[12] = local_radius; o[13] = density; o[14] = mean_curv; o[15] = dir_cons;
  o[16] = mx; o[17] = my; o[18] = mz;
  o[19] = stdx; o[20] = stdy; o[21] = stdz;
  o[22] = 0.f; o[23] = 0.f;
}

// ---------------------------------------------------------------------------
// Kernel C2: structure MLP over 16-point tiles (22 padded to 24 -> 6 WMMAs),
// then BN+ReLU, then 16x16 layer (4 WMMAs). Output channels 16..31.
// ---------------------------------------------------------------------------
__global__ void struct_mlp_kernel(const float* __restrict__ sf,
                                  const float* __restrict__ W1,
                                  const float* __restrict__ b1,
                                  const float* __restrict__ g,
                                  const float* __restrict__ be,
                                  const float* __restrict__ m,
                                  const float* __restrict__ v,
                                  const float* __restrict__ W2,
                                  const float* __restrict__ b2,
                                  float* __restrict__ out) {
  __shared__ float hlds[8][16][17];
  const int lane = threadIdx.x & 31;
  const int wave = threadIdx.x >> 5;
  const int half = lane >> 4;
  const int mrow = lane & 15;
  const int pbase = (blockIdx.x * 8 + wave) * 16;  // 16 points per wave

  // Layer 1: D = sf(16x24) x W1s^T(24x16), 6 WMMAs (cols 22,23 are zero).
  v8f acc = {0.f, 0.f, 0.f, 0.f, 0.f, 0.f, 0.f, 0.f};
#pragma unroll
  for (int kk = 0; kk < 6; ++kk) {
    const int k0 = 4 * kk + 2 * half;
    const float* row = sf + (size_t)(pbase + mrow) * 24;
    v2f a = {row[k0], row[k0 + 1]};
    v2f bf = {(k0 < 22 ? W1[mrow * 22 + k0] : 0.f),
              (k0 + 1 < 22 ? W1[mrow * 22 + k0 + 1] : 0.f)};
    acc = wmma4(a, bf, acc);
  }

  const float sc = g[mrow] * rsqrtf(v[mrow] + 1e-5f);
  const float tt = (b1[mrow] - m[mrow]) * sc + be[mrow];
#pragma unroll
  for (int j = 0; j < 8; ++j) {
    const float x = acc[j] * sc + tt;
    hlds[wave][j + 8 * half][mrow] = x > 0.f ? x : 0.f;
  }
  __syncthreads();

  v8f acc2 = {0.f, 0.f, 0.f, 0.f, 0.f, 0.f, 0.f, 0.f};
#pragma unroll
  for (int kk = 0; kk < 4; ++kk) {
    const int k0 = 4 * kk + 2 * half;
    v2f a = {hlds[wave][mrow][k0], hlds[wave][mrow][k0 + 1]};
    v2f bf = {W2[mrow * 16 + k0], W2[mrow * 16 + k0 + 1]};
    acc2 = wmma4(a, bf, acc2);
  }

  // Lane holds column (=channel) mrow for point rows j+8*half.
#pragma unroll
  for (int j = 0; j < 8; ++j) {
    const int pt = pbase + j + 8 * half;
    const int bb = pt >> 12, nn = pt & (NP - 1);
    out[((size_t)(bb * 32 + 16 + mrow)) * NP + nn] = acc2[j] + b2[mrow];
  }
}

// ---------------------------------------------------------------------------
extern "C" void kernel_launch(void* const* d_in, const int* in_sizes, int n_in,
                              void* d_out, int out_size, void* d_ws,
                              size_t ws_size, hipStream_t stream) {
  (void)in_sizes; (void)n_in; (void)out_size; (void)ws_size;
  const float* xyz  = (const float*)d_in[0];
  const float* W1r  = (const float*)d_in[1];
  const float* b1r  = (const float*)d_in[2];
  const float* g_r  = (const float*)d_in[3];
  const float* be_r = (const float*)d_in[4];
  const float* m_r  = (const float*)d_in[5];
  const float* v_r  = (const float*)d_in[6];
  const float* W2r  = (const float*)d_in[7];
  const float* b2r  = (const float*)d_in[8];
  const float* W1s  = (const float*)d_in[9];
  const float* b1s  = (const float*)d_in[10];
  const float* g_s  = (const float*)d_in[11];
  const float* be_s = (const float*)d_in[12];
  const float* m_s  = (const float*)d_in[13];
  const float* v_s  = (const float*)d_in[14];
  const float* W2s  = (const float*)d_in[15];
  const float* b2s  = (const float*)d_in[16];
  float* out = (float*)d_out;

  float* relpos = (float*)d_ws;                       // 4*4096*16*3 floats = 3 MB
  float* sf = relpos + (size_t)NB * NP * KNN * 3;     // 4*4096*24 floats  = 1.5 MB

  knn_relpos_kernel<<<NB * (NP / TILE), TILE, 0, stream>>>(xyz, relpos);
  rel_mlp_kernel<<<(NB * NP) / 8, 256, 0, stream>>>(
      relpos, W1r, b1r, g_r, be_r, m_r, v_r, W2r, b2r, out);
  struct_feat_kernel<<<NB * (NP / 256), 256, 0, stream>>>(relpos, sf);
  struct_mlp_kernel<<<(NB * NP) / (8 * 16), 256, 0, stream>>>(
      sf, W1s, b1s, g_s, be_s, m_s, v_s, W2s, b2s, out);
}